// DynamicRouting_49366354100543
// MI455X (gfx1250) — compile-verified
//
#include <hip/hip_runtime.h>

// Dynamic capsule routing, MI455X (gfx1250, wave32).
// B=128 batches, J=32 parent caps, I=1152 child caps, N=16 dim, iters=3,
// sparse top-k masking 12 edges per (b,i) column at update iters 0 and 1.

#define NJ 32
#define NI 1152
#define NN 16
#define NB 128
#define MASKK 12
#define SJN (NJ * NN) // 512

typedef __attribute__((ext_vector_type(2))) float v2f;
typedef __attribute__((ext_vector_type(8))) float v8f;

// ---------------------------------------------------------------------------
// Fused per-iteration pass.
// One wave = 32 consecutive i-columns of one batch b. Lane = i column.
// MODE 0: c = 1/32 uniform (iteration 0, b == 0), only s-partials.
// MODE 1: b = agreement(u, v_in); sparsify(12); softmax -> c; s-partials.
// MODE 2: b = b_prev + agreement(u, v_in); sparsify(12); softmax; s-partials.
// s-partials use V_WMMA_F32_16X16X4_F32 as a cross-lane (K = i) reduction:
//   A[m][k] = c[j][i_{4w+k}]  (broadcast over m, from LDS c-table)
//   B[k][n] = u_hat[b][j][i_{4w+k}][n]  (n across lanes)
//   D row 0, lane n (<16)  ->  partial s[j][n] over the wave's 32 i.
// ---------------------------------------------------------------------------
template <int MODE>
__global__ __launch_bounds__(128) void routing_pass(
    const float* __restrict__ u_hat, const float* __restrict__ v_in,
    float* __restrict__ b_vec, float* __restrict__ s_out) {
  __shared__ float v_lds[SJN];       // v[j][n] staged for the block
  __shared__ float s_lds[SJN];       // block-level s accumulator
  __shared__ float c_lds[128 * NJ];  // c[j] per thread-column (16 KB)

  const int tid = threadIdx.x;
  const int b = blockIdx.y;
  const int wave = tid >> 5;
  const int lane = tid & 31;
  const int i_base = (blockIdx.x * 4 + wave) * 32;
  const int i = i_base + lane;
  const float NEG_INF = -__builtin_inff();
  const float POS_INF = __builtin_inff();

  for (int t = tid; t < SJN; t += 128) {
    s_lds[t] = 0.0f;
    if (MODE != 0) v_lds[t] = v_in[(size_t)b * SJN + t];
  }
  __syncthreads();

  float c_arr[NJ];

  if (MODE == 0) {
#pragma unroll
    for (int j = 0; j < NJ; ++j) c_arr[j] = 1.0f / 32.0f;
  } else {
    float bcol[NJ];
    // ---- sweep A: agreement (in-lane dot over n) + b update ----
#pragma unroll
    for (int j = 0; j < NJ; ++j) {
      const float* up = u_hat + (((size_t)b * NJ + j) * NI + i) * NN;
      const float4* u4 = reinterpret_cast<const float4*>(up);
      float4 u0 = u4[0], u1 = u4[1], u2 = u4[2], u3 = u4[3];
      const float4* w4 = reinterpret_cast<const float4*>(v_lds + j * NN);
      float4 w0 = w4[0], w1 = w4[1], w2 = w4[2], w3 = w4[3];
      float a = u0.x * w0.x + u0.y * w0.y + u0.z * w0.z + u0.w * w0.w +
                u1.x * w1.x + u1.y * w1.y + u1.z * w1.z + u1.w * w1.w +
                u2.x * w2.x + u2.y * w2.y + u2.z * w2.z + u2.w * w2.w +
                u3.x * w3.x + u3.y * w3.y + u3.z * w3.z + u3.w * w3.w;
      float bprev =
          (MODE == 2) ? b_vec[((size_t)b * NJ + j) * NI + i] : 0.0f;
      bcol[j] = bprev + a;  // -inf + finite stays -inf
    }
    // ---- sparsify: mask the MASKK smallest (prev -inf treated as +inf) ----
    {
      float bt[NJ];
#pragma unroll
      for (int j = 0; j < NJ; ++j)
        bt[j] = (bcol[j] == NEG_INF) ? POS_INF : bcol[j];
      float cur = NEG_INF;
#pragma unroll
      for (int t = 0; t < MASKK; ++t) {  // 12-step dup-skip selection
        float m = POS_INF;
#pragma unroll
        for (int j = 0; j < NJ; ++j)
          if (bt[j] > cur) m = fminf(m, bt[j]);
        cur = m;
      }
      const float kth = cur;
#pragma unroll
      for (int j = 0; j < NJ; ++j)
        if (bt[j] <= kth || bcol[j] == NEG_INF) bcol[j] = NEG_INF;
    }
    // ---- write b state, masked softmax over j ----
    float mmax = NEG_INF;
#pragma unroll
    for (int j = 0; j < NJ; ++j) {
      b_vec[((size_t)b * NJ + j) * NI + i] = bcol[j];
      mmax = fmaxf(mmax, bcol[j]);
    }
    float sum = 0.0f;
#pragma unroll
    for (int j = 0; j < NJ; ++j) {
      float e = __expf(bcol[j] - mmax);  // exp(-inf)=0
      c_arr[j] = e;
      sum += e;
    }
    const float inv = 1.0f / sum;
#pragma unroll
    for (int j = 0; j < NJ; ++j) c_arr[j] *= inv;
  }

  // publish c so every lane can fetch any column's c[j] for the A operand
#pragma unroll
  for (int j = 0; j < NJ; ++j) c_lds[tid * NJ + j] = c_arr[j];
  __syncthreads();

  // ---- sweep B: WMMA s-partials (cross-lane reduction over i) ----
  const int half = (lane < 16) ? 0 : 2;  // K rows held by this half-wave
  const int ncol = lane & 15;            // B/D column = n
  const int crow = wave * 32;            // this wave's c-table base

  for (int j = 0; j < NJ; ++j) {
    const float* ub = u_hat + (((size_t)b * NJ + j) * NI + i_base) * NN;
    v8f acc = {0.f, 0.f, 0.f, 0.f, 0.f, 0.f, 0.f, 0.f};
#pragma unroll
    for (int w = 0; w < 8; ++w) {  // 8 x (K=4) covers the wave's 32 i
      const int i0 = 4 * w + half;
      v2f aop, bop;
      aop.x = c_lds[(crow + i0) * NJ + j];
      aop.y = c_lds[(crow + i0 + 1) * NJ + j];
      bop.x = ub[i0 * NN + ncol];
      bop.y = ub[(i0 + 1) * NN + ncol];
      acc = __builtin_amdgcn_wmma_f32_16x16x4_f32(
          false, aop, false, bop, (short)0, acc, false, false);
    }
    if (lane < 16) atomicAdd(&s_lds[j * NN + ncol], acc.s0);
  }
  __syncthreads();
  for (int t = tid; t < SJN; t += 128)
    atomicAdd(&s_out[(size_t)b * SJN + t], s_lds[t]);
}

// ---------------------------------------------------------------------------
__global__ __launch_bounds__(256) void squash_kernel(
    const float* __restrict__ s, float* __restrict__ v) {
  const int idx = blockIdx.x * 256 + threadIdx.x;  // (b*J + j)*16 + n
  float sv = s[idx];
  float sq = sv * sv;
#pragma unroll
  for (int m = 1; m < 16; m <<= 1) sq += __shfl_xor(sq, m, 16);
  const float scale = (sq / (1.0f + sq)) * rsqrtf(sq + 1e-8f);
  v[idx] = sv * scale;
}

__global__ __launch_bounds__(256) void zero_kernel(float* __restrict__ p,
                                                   int n) {
  const int i = blockIdx.x * 256 + threadIdx.x;
  if (i < n) p[i] = 0.0f;
}

// ---------------------------------------------------------------------------
extern "C" void kernel_launch(void* const* d_in, const int* in_sizes, int n_in,
                              void* d_out, int out_size, void* d_ws,
                              size_t ws_size, hipStream_t stream) {
  (void)in_sizes; (void)n_in; (void)out_size; (void)ws_size;
  const float* u_hat = (const float*)d_in[0];  // (128,32,1152,16) fp32
  float* ws = (float*)d_ws;
  float* s = ws;                              // 65536 floats
  float* v = ws + (NB * SJN);                 // 65536 floats
  float* bvec = ws + 2 * (NB * SJN);          // 128*32*1152 floats
  float* out = (float*)d_out;

  const dim3 pg(NI / 128, NB), pb(128);
  const dim3 zg((NB * SJN + 255) / 256), zb(256);
  const dim3 qg((NB * SJN) / 256), qb(256);

  // iter 0: uniform c -> s0 -> v0
  zero_kernel<<<zg, zb, 0, stream>>>(s, NB * SJN);
  routing_pass<0><<<pg, pb, 0, stream>>>(u_hat, nullptr, bvec, s);
  squash_kernel<<<qg, qb, 0, stream>>>(s, v);
  // iter 1: b1 = agree(u, v0); sparsify 12; c1 -> s1 -> v1
  zero_kernel<<<zg, zb, 0, stream>>>(s, NB * SJN);
  routing_pass<1><<<pg, pb, 0, stream>>>(u_hat, v, bvec, s);
  squash_kernel<<<qg, qb, 0, stream>>>(s, v);
  // iter 2: b2 = b1 + agree(u, v1); sparsify 12; c2 -> s2 -> v2 = output
  zero_kernel<<<zg, zb, 0, stream>>>(s, NB * SJN);
  routing_pass<2><<<pg, pb, 0, stream>>>(u_hat, v, bvec, s);
  squash_kernel<<<qg, qb, 0, stream>>>(s, out);
}